// FeatureMapApproximator_38878043963448
// MI455X (gfx1250) — compile-verified
//
#include <hip/hip_runtime.h>
#include <math.h>

#define FDIM   512
#define LPIX   4096
#define DINNER 1024
#define DSTATE 16
#define DTRANK 32
#define KDIR   4

typedef float v2f __attribute__((ext_vector_type(2)));
typedef float v8f __attribute__((ext_vector_type(8)));

__device__ __forceinline__ float geluf(float x) {
  return 0.5f * x * (1.0f + erff(x * 0.70710678118654752f));
}
__device__ __forceinline__ float siluf(float x) {
  return x / (1.0f + expf(-x));
}
__device__ __forceinline__ float softplusf(float x) {
  return (x > 20.0f) ? x : log1pf(expf(x));
}

// ---------------------------------------------------------------------------
// Generic WMMA f32 GEMM: C[M,N] = act(A[M,K] * B'[K,N] + bias) + addsrc
//  A row-major (weights), B' per BMODE:
//   BMODE 0: B'[k][n] = B[k*N+n]
//   BMODE 2: channel shuffle (groups=8 over 512 ch): B'[k] = B[(k%8)*64 + k/8]
//  act: 0 none, 1 gelu, 3 silu
// Block: 128 threads (4 waves), tile 64(M) x 64(N); wave w -> rows [16w,16w+16)
// ---------------------------------------------------------------------------
template <int BMODE>
__global__ __launch_bounds__(128)
void k_gemm_wmma(const float* __restrict__ A, const float* __restrict__ B,
                 float* __restrict__ C, int M, int N, int Kd,
                 const float* __restrict__ bias,
                 const float* __restrict__ addsrc, int act)
{
  const int lane  = threadIdx.x & 31;
  const int wave  = threadIdx.x >> 5;
  const int l16   = lane & 15;
  const int khalf = (lane >> 4) << 1;           // 0 or 2 (ISA f32 A/B layout)
  const int nb    = blockIdx.x * 64;
  const int mb    = blockIdx.y * 64 + wave * 16;
  const int mrow  = mb + l16;
  const size_t arow = (size_t)mrow * Kd;

  v8f acc[4] = {};

  for (int k0 = 0; k0 < Kd; k0 += 4) {
    const int kk0 = k0 + khalf;
    v2f a = *reinterpret_cast<const v2f*>(&A[arow + kk0]);   // 8B aligned
#pragma unroll
    for (int nt = 0; nt < 4; ++nt) {
      const int ncol = nb + nt * 16 + l16;
      v2f b;
      if (BMODE == 0) {
        b[0] = B[(size_t)(kk0 + 0) * N + ncol];
        b[1] = B[(size_t)(kk0 + 1) * N + ncol];
      } else { // BMODE 2: channel shuffle, groups=8 over 512 channels
#pragma unroll
        for (int e = 0; e < 2; ++e) {
          const int kk  = kk0 + e;
          const int src = ((kk & 7) << 6) + (kk >> 3);
          b[e] = B[(size_t)src * N + ncol];
        }
      }
      acc[nt] = __builtin_amdgcn_wmma_f32_16x16x4_f32(
          false, a, false, b, (short)0, acc[nt], false, false);
    }
  }

  const int moff = (lane >> 4) ? 8 : 0;
#pragma unroll
  for (int nt = 0; nt < 4; ++nt) {
    const int ncol = nb + nt * 16 + l16;
#pragma unroll
    for (int r = 0; r < 8; ++r) {
      const int m = mb + r + moff;
      float v = acc[nt][r];
      if (bias) v += bias[m];
      if (act == 1) v = geluf(v);
      else if (act == 3) v = siluf(v);
      const size_t off = (size_t)m * N + ncol;
      if (addsrc) v += addsrc[off];
      C[off] = v;
    }
  }
}

// ---------------------------------------------------------------------------
// Implicit-GEMM 3x3 conv (pad=1, dil=1), tap-outer / channel-inner.
// A = weights (M x Cin*9) with k = c*9 + t;  B = input (Cin x 64x64).
// Block tile 64(M) x 64(N); the 64 N-columns span exactly one image row, so
// the row bound is scalar and per-tap offsets/masks are hoisted out of the
// 1024-deep channel loop: inner loop = 2 A loads + 8 masked B loads + 4 WMMA.
// ---------------------------------------------------------------------------
__global__ __launch_bounds__(128)
void k_conv3x3_wmma(const float* __restrict__ A, const float* __restrict__ B,
                    float* __restrict__ C, int M, int Cin)
{
  const int lane  = threadIdx.x & 31;
  const int wave  = threadIdx.x >> 5;
  const int l16   = lane & 15;
  const int khalf = (lane >> 4) << 1;
  const int nb    = blockIdx.x * 64;
  const int mb    = blockIdx.y * 64 + wave * 16;
  const int mrow  = mb + l16;
  const int py    = nb >> 6;                    // block-uniform image row
  const size_t arow = (size_t)mrow * (size_t)(Cin * 9);

  v8f acc[4] = {};

  for (int t = 0; t < 9; ++t) {
    const int dy = t / 3 - 1, dx = t % 3 - 1;   // constants after unswitching
    const int yy = py + dy;
    const bool oky = (unsigned)yy < 64u;
    int   poff[4];
    float okf[4];
#pragma unroll
    for (int nt = 0; nt < 4; ++nt) {
      const int xx = nt * 16 + l16 + dx;
      const bool ok = oky && ((unsigned)xx < 64u);
      poff[nt] = ok ? ((yy << 6) + xx) : 0;     // clamped: always in-bounds
      okf[nt]  = ok ? 1.0f : 0.0f;
    }
    for (int c0 = 0; c0 < Cin; c0 += 4) {
      const int kk0 = c0 + khalf;
      v2f a;
      a[0] = A[arow + (size_t)(kk0 + 0) * 9 + t];
      a[1] = A[arow + (size_t)(kk0 + 1) * 9 + t];
#pragma unroll
      for (int nt = 0; nt < 4; ++nt) {
        v2f b;
        b[0] = B[(size_t)(kk0 + 0) * LPIX + poff[nt]] * okf[nt];
        b[1] = B[(size_t)(kk0 + 1) * LPIX + poff[nt]] * okf[nt];
        acc[nt] = __builtin_amdgcn_wmma_f32_16x16x4_f32(
            false, a, false, b, (short)0, acc[nt], false, false);
      }
    }
  }

  const int moff = (lane >> 4) ? 8 : 0;
#pragma unroll
  for (int nt = 0; nt < 4; ++nt) {
    const int ncol = nb + nt * 16 + l16;
#pragma unroll
    for (int r = 0; r < 8; ++r) {
      const int m = mb + r + moff;
      C[(size_t)m * LPIX + ncol] = acc[nt][r];
    }
  }
}

// ---------------------------------------------------------------------------
// Fusion gate: g = sigmoid(W2*gelu(W1*concat+b1)+b2); wf = concat*g
// One block per pixel; concat = [feature; latent] (1024 ch).
// ---------------------------------------------------------------------------
__global__ __launch_bounds__(256)
void k_fusegate(const float* __restrict__ feat, const float* __restrict__ lat,
                const float* __restrict__ w1, const float* __restrict__ b1,
                const float* __restrict__ w2, const float* __restrict__ b2,
                float* __restrict__ wf)
{
  __shared__ float cbuf[2 * FDIM];
  __shared__ float red[256];
  __shared__ float mid[32];
  const int p = blockIdx.x;
  const int t = threadIdx.x;
#pragma unroll
  for (int i = 0; i < 4; ++i) {
    const int c = t + i * 256;
    cbuf[c] = (c < FDIM) ? feat[(size_t)c * LPIX + p]
                         : lat[(size_t)(c - FDIM) * LPIX + p];
  }
  __syncthreads();
  {
    const int m = t >> 3, part = t & 7;
    float s = 0.0f;
    const float* wr = w1 + (size_t)m * (2 * FDIM) + part * 128;
    const float* cb = cbuf + part * 128;
    for (int i = 0; i < 128; ++i) s += wr[i] * cb[i];
    red[t] = s;
  }
  __syncthreads();
  if ((t & 7) == 0) {
    float s = 0.0f;
    for (int i = 0; i < 8; ++i) s += red[t + i];
    mid[t >> 3] = geluf(s + b1[t >> 3]);
  }
  __syncthreads();
#pragma unroll
  for (int i = 0; i < 4; ++i) {
    const int c = t + i * 256;
    float s = b2[c];
    const float* wr = w2 + (size_t)c * 32;
#pragma unroll
    for (int m = 0; m < 32; ++m) s += wr[m] * mid[m];
    const float g = 1.0f / (1.0f + expf(-s));
    wf[(size_t)c * LPIX + p] = cbuf[c] * g;
  }
}

// Whole-tensor sum / sum-of-squares reduction (for LN over axes 1,2,3)
__global__ __launch_bounds__(256)
void k_reduce_sumsq(const float* __restrict__ X, size_t n, float* __restrict__ stats)
{
  __shared__ float s1[256], s2[256];
  size_t i = (size_t)blockIdx.x * 256 + threadIdx.x;
  const size_t stride = (size_t)gridDim.x * 256;
  float a = 0.0f, q = 0.0f;
  for (; i < n; i += stride) { const float v = X[i]; a += v; q += v * v; }
  s1[threadIdx.x] = a; s2[threadIdx.x] = q;
  __syncthreads();
  for (int o = 128; o > 0; o >>= 1) {
    if ((int)threadIdx.x < o) { s1[threadIdx.x] += s1[threadIdx.x + o];
                                s2[threadIdx.x] += s2[threadIdx.x + o]; }
    __syncthreads();
  }
  if (threadIdx.x == 0) { atomicAdd(&stats[0], s1[0]); atomicAdd(&stats[1], s2[0]); }
}

__global__ __launch_bounds__(256)
void k_ln_global_apply(const float* __restrict__ X, const float* __restrict__ g,
                       const float* __restrict__ b, const float* __restrict__ stats,
                       float* __restrict__ Y, size_t n)
{
  const size_t i = (size_t)blockIdx.x * 256 + threadIdx.x;
  if (i >= n) return;
  const float mu  = stats[0] / (float)n;
  const float var = stats[1] / (float)n - mu * mu;
  const float inv = rsqrtf(var + 1e-5f);
  const float v = (X[i] - mu) * inv * g[i] + b[i];
  Y[i] = geluf(v);
}

// Per-pixel LN over C channels; optional SiLU(z) gate fused.
__global__ __launch_bounds__(256)
void k_pixel_ln(const float* __restrict__ X, const float* __restrict__ g,
                const float* __restrict__ b, const float* __restrict__ z,
                float* __restrict__ Y, int C)
{
  __shared__ float s1[256], s2[256];
  const int p = blockIdx.x, t = threadIdx.x;
  float a = 0.0f, q = 0.0f;
  for (int c = t; c < C; c += 256) {
    const float v = X[(size_t)c * LPIX + p]; a += v; q += v * v;
  }
  s1[t] = a; s2[t] = q;
  __syncthreads();
  for (int o = 128; o > 0; o >>= 1) {
    if (t < o) { s1[t] += s1[t + o]; s2[t] += s2[t + o]; }
    __syncthreads();
  }
  const float mu  = s1[0] / (float)C;
  const float var = s2[0] / (float)C - mu * mu;
  const float inv = rsqrtf(var + 1e-5f);
  for (int c = t; c < C; c += 256) {
    const size_t off = (size_t)c * LPIX + p;
    float v = (X[off] - mu) * inv * g[c] + b[c];
    if (z) v *= siluf(z[off]);
    Y[off] = v;
  }
}

// Depthwise 3x3 conv, pad = dil, optional SiLU.
__global__ __launch_bounds__(256)
void k_dwconv(const float* __restrict__ X, const float* __restrict__ Wd,
              const float* __restrict__ bias, float* __restrict__ Y,
              int C, int dil, int act)
{
  const int i = blockIdx.x * 256 + threadIdx.x;
  if (i >= C * LPIX) return;
  const int c = i >> 12, p = i & 4095;
  const int py = p >> 6, px = p & 63;
  float s = bias ? bias[c] : 0.0f;
#pragma unroll
  for (int t = 0; t < 9; ++t) {
    const int yy = py + ((t / 3) - 1) * dil;
    const int xx = px + ((t % 3) - 1) * dil;
    if ((unsigned)yy < 64u && (unsigned)xx < 64u)
      s += Wd[c * 9 + t] * X[(size_t)c * LPIX + (yy << 6) + xx];
  }
  if (act == 3) s = siluf(s);
  Y[i] = s;
}

// Build scan directions 1..3 from direction 0 (xc) already at xs slice 0.
__global__ __launch_bounds__(256)
void k_build_xs(const float* __restrict__ xc, float* __restrict__ xs)
{
  const int i = blockIdx.x * 256 + threadIdx.x;
  if (i >= DINNER * LPIX) return;
  const int d = i >> 12, l = i & 4095;
  const int lt  = ((l & 63) << 6) + (l >> 6);
  const int lf  = 4095 - l;
  const int lft = ((lf & 63) << 6) + (lf >> 6);
  const size_t row  = (size_t)d * LPIX;
  const size_t base = row + l;
  const size_t S = (size_t)DINNER * LPIX;
  xs[1 * S + base] = xc[row + lt];
  xs[2 * S + base] = xc[row + lf];
  xs[3 * S + base] = xc[row + lft];
}

// Selective scan. Block = 128 threads = 8 channels x 16 states; lane n = state.
// grid = KDIR * (DINNER/8). Accumulates remapped y via f32 atomics.
__global__ __launch_bounds__(128)
void k_scan(const float* __restrict__ xs, const float* __restrict__ xdbl,
            const float* __restrict__ draw, const float* __restrict__ A_logs,
            const float* __restrict__ Ds, float* __restrict__ y)
{
  const int blk  = blockIdx.x;
  const int k    = blk >> 7;
  const int dblk = blk & 127;
  const int t    = threadIdx.x;
  const int dl   = t >> 4, n = t & 15;
  const int d    = dblk * 8 + dl;

  const float* up = xs   + ((size_t)k * DINNER + d) * LPIX;
  const float* dp = draw + ((size_t)k * DINNER + d) * LPIX;
  const float* Bp = xdbl + (size_t)(k * 64 + DTRANK + n) * LPIX;
  const float* Cp = xdbl + (size_t)(k * 64 + DTRANK + DSTATE + n) * LPIX;
  const float Aval = -expf(A_logs[((size_t)k * DINNER + d) * DSTATE + n]);
  const float Dval = Ds[k * DINNER + d];

  float h = 0.0f;
  for (int l = 0; l < LPIX; ++l) {
    const float delta = softplusf(dp[l]);
    const float u  = up[l];
    const float dA = expf(delta * Aval);
    h = dA * h + (delta * u) * Bp[l];
    float yl = h * Cp[l];
    yl += __shfl_xor(yl, 8);
    yl += __shfl_xor(yl, 4);
    yl += __shfl_xor(yl, 2);
    yl += __shfl_xor(yl, 1);
    if (n == 0) {
      yl += Dval * u;
      int p;
      if (k == 0)      p = l;
      else if (k == 1) p = ((l & 63) << 6) + (l >> 6);
      else if (k == 2) p = 4095 - l;
      else { const int lf = 4095 - l; p = ((lf & 63) << 6) + (lf >> 6); }
      atomicAdd(&y[(size_t)d * LPIX + p], yl);
    }
  }
}

// 3x3 box blur (zero-pad, /9) + residual.
__global__ __launch_bounds__(256)
void k_blur_add(const float* __restrict__ attn, const float* __restrict__ fused,
                float* __restrict__ out)
{
  const int i = blockIdx.x * 256 + threadIdx.x;
  if (i >= FDIM * LPIX) return;
  const int c = i >> 12, p = i & 4095;
  const int py = p >> 6, px = p & 63;
  float s = 0.0f;
#pragma unroll
  for (int t = 0; t < 9; ++t) {
    const int yy = py + t / 3 - 1, xx = px + t % 3 - 1;
    if ((unsigned)yy < 64u && (unsigned)xx < 64u)
      s += attn[(size_t)c * LPIX + (yy << 6) + xx];
  }
  out[i] = s * (1.0f / 9.0f) + fused[i];
}

// ---------------------------------------------------------------------------
extern "C" void kernel_launch(void* const* d_in, const int* in_sizes, int n_in,
                              void* d_out, int out_size, void* d_ws, size_t ws_size,
                              hipStream_t stream)
{
  (void)in_sizes; (void)n_in; (void)out_size; (void)ws_size;
  const float* feat       = (const float*)d_in[0];
  const float* lat        = (const float*)d_in[1];
  const float* fg_w1      = (const float*)d_in[2];
  const float* fg_b1      = (const float*)d_in[3];
  const float* fg_w2      = (const float*)d_in[4];
  const float* fg_b2      = (const float*)d_in[5];
  const float* fb_w1      = (const float*)d_in[6];
  const float* ln_g       = (const float*)d_in[7];
  const float* ln_b       = (const float*)d_in[8];
  const float* fb_dw_w    = (const float*)d_in[9];
  const float* fb_dw_b    = (const float*)d_in[10];
  const float* fb_pw1_w   = (const float*)d_in[11];
  const float* fb_pw1_b   = (const float*)d_in[12];
  const float* fb_pw2_w   = (const float*)d_in[13];
  const float* fb_pw2_b   = (const float*)d_in[14];
  const float* ln1_g      = (const float*)d_in[15];
  const float* ln1_b      = (const float*)d_in[16];
  const float* in_proj_w  = (const float*)d_in[17];
  const float* dconv_w    = (const float*)d_in[18];
  const float* dconv_b    = (const float*)d_in[19];
  const float* x_proj_w   = (const float*)d_in[20];
  const float* dt_w       = (const float*)d_in[21];
  const float* dt_b       = (const float*)d_in[22];
  const float* A_logs     = (const float*)d_in[23];
  const float* Ds         = (const float*)d_in[24];
  const float* out_norm_g = (const float*)d_in[25];
  const float* out_norm_b = (const float*)d_in[26];
  const float* out_proj_w = (const float*)d_in[27];

  float* ws = (float*)d_ws;
  const size_t DL = (size_t)DINNER * LPIX;   // 4,194,304
  const size_t FL = (size_t)FDIM * LPIX;     // 2,097,152
  const size_t o_wf    = 0;                  // wf; later reused for y accum
  const size_t o_h1    = o_wf + DL;
  const size_t o_h2    = o_h1 + FL;
  const size_t o_fused = o_h2 + FL;
  const size_t o_xz    = o_fused + FL;
  const size_t o_xs    = o_xz + 2 * DL;
  const size_t o_xdbl  = o_xs + KDIR * DL;
  const size_t o_draw  = o_xdbl + (size_t)KDIR * 64 * LPIX; // later: ygated
  const size_t o_stats = o_draw + KDIR * DL;

  const dim3 b128(128), b256(256);

  // 1. fusion gate -> wf
  k_fusegate<<<LPIX, b256, 0, stream>>>(feat, lat, fg_w1, fg_b1, fg_w2, fg_b2,
                                        ws + o_wf);
  // 2. 3x3 conv 1024->512 as implicit WMMA GEMM -> h1
  {
    dim3 g(LPIX / 64, FDIM / 64);
    k_conv3x3_wmma<<<g, b128, 0, stream>>>(fb_w1, ws + o_wf, ws + o_h1,
                                           FDIM, DINNER);
  }
  // 3. global LN + gelu: h1 -> h2
  hipMemsetAsync(ws + o_stats, 0, 2 * sizeof(float), stream);
  k_reduce_sumsq<<<1024, b256, 0, stream>>>(ws + o_h1, FL, ws + o_stats);
  k_ln_global_apply<<<(unsigned)((FL + 255) / 256), b256, 0, stream>>>(
      ws + o_h1, ln_g, ln_b, ws + o_stats, ws + o_h2, FL);
  // 4. depthwise dil=2 conv: h2 -> h1
  k_dwconv<<<(FDIM * LPIX + 255) / 256, b256, 0, stream>>>(
      ws + o_h2, fb_dw_w, fb_dw_b, ws + o_h1, FDIM, 2, 0);
  // 5. pw1 + gelu: h1 -> h2
  {
    dim3 g(LPIX / 64, FDIM / 64);
    k_gemm_wmma<0><<<g, b128, 0, stream>>>(fb_pw1_w, ws + o_h1, ws + o_h2,
        FDIM, LPIX, FDIM, fb_pw1_b, nullptr, 1);
  }
  // 6. channel-shuffle + pw2 + residual(feature) -> fused
  {
    dim3 g(LPIX / 64, FDIM / 64);
    k_gemm_wmma<2><<<g, b128, 0, stream>>>(fb_pw2_w, ws + o_h2, ws + o_fused,
        FDIM, LPIX, FDIM, fb_pw2_b, feat, 0);
  }
  // 7. per-pixel LN (ln1): fused -> xn (h1)
  k_pixel_ln<<<LPIX, b256, 0, stream>>>(ws + o_fused, ln1_g, ln1_b, nullptr,
                                        ws + o_h1, FDIM);
  // 8. in_proj: xn -> xz (2048 x L)
  {
    dim3 g(LPIX / 64, (2 * DINNER) / 64);
    k_gemm_wmma<0><<<g, b128, 0, stream>>>(in_proj_w, ws + o_h1, ws + o_xz,
        2 * DINNER, LPIX, FDIM, nullptr, nullptr, 0);
  }
  // 9. depthwise conv + silu on xp -> xs slice 0
  k_dwconv<<<(DINNER * LPIX + 255) / 256, b256, 0, stream>>>(
      ws + o_xz, dconv_w, dconv_b, ws + o_xs, DINNER, 1, 3);
  // 10. build scan directions 1..3
  k_build_xs<<<(DINNER * LPIX + 255) / 256, b256, 0, stream>>>(
      ws + o_xs, ws + o_xs);
  // 11. x_proj per direction: (64 x 1024) @ (1024 x L) -> xdbl
  for (int k = 0; k < KDIR; ++k) {
    dim3 g(LPIX / 64, 1);
    k_gemm_wmma<0><<<g, b128, 0, stream>>>(
        x_proj_w + (size_t)k * 64 * DINNER,
        ws + o_xs + (size_t)k * DL,
        ws + o_xdbl + (size_t)k * 64 * LPIX,
        64, LPIX, DINNER, nullptr, nullptr, 0);
  }
  // 12. dt projection per direction (+dt_b): (1024 x 32) @ dts -> draw
  for (int k = 0; k < KDIR; ++k) {
    dim3 g(LPIX / 64, DINNER / 64);
    k_gemm_wmma<0><<<g, b128, 0, stream>>>(
        dt_w + (size_t)k * DINNER * DTRANK,
        ws + o_xdbl + (size_t)k * 64 * LPIX,   // rows 0..31 = dts
        ws + o_draw + (size_t)k * DL,
        DINNER, LPIX, DTRANK, dt_b + (size_t)k * DINNER, nullptr, 0);
  }
  // 13. selective scan -> combined y (reuse o_wf)
  hipMemsetAsync(ws + o_wf, 0, DL * sizeof(float), stream);
  k_scan<<<KDIR * (DINNER / 8), b128, 0, stream>>>(
      ws + o_xs, ws + o_xdbl, ws + o_draw, A_logs, Ds, ws + o_wf);
  // 14. out_norm per-pixel LN + silu(z) gate -> ygated (reuse o_draw)
  k_pixel_ln<<<LPIX, b256, 0, stream>>>(ws + o_wf, out_norm_g, out_norm_b,
                                        ws + o_xz + DL, ws + o_draw, DINNER);
  // 15. out_proj + residual(fused) -> attn (h1)
  {
    dim3 g(LPIX / 64, FDIM / 64);
    k_gemm_wmma<0><<<g, b128, 0, stream>>>(out_proj_w, ws + o_draw, ws + o_h1,
        FDIM, LPIX, DINNER, nullptr, ws + o_fused, 0);
  }
  // 16. 3x3 box blur + residual(fused) -> output
  k_blur_add<<<(FDIM * LPIX + 255) / 256, b256, 0, stream>>>(
      ws + o_h1, ws + o_fused, (float*)d_out);
}